// SupervisedTree_4947802325040
// MI455X (gfx1250) — compile-verified
//
#include <hip/hip_runtime.h>
#include <hip/hip_bf16.h>

// ---------------------------------------------------------------------------
// SupervisedTree distance kernel for MI455X (gfx1250, wave32, WMMA)
//
//   B    = softmax(param, axis=0)                      (625, 4096)
//   psub = subtree-sum(B)  (== (I - D1)^{-1} @ B)      (625, 4096)
//   z    = mass @ psub.T                               (128, 625)   <- WMMA f32
//   out  = sum_i x*tanh(x) over z-dims + mass-dims     (128, 128)
// ---------------------------------------------------------------------------

#define N_INNER   625
#define N_INNER_P 640      // padded to multiple of 16 for WMMA tiles
#define N_LEAF    4096
#define N_DOC     128

typedef __attribute__((ext_vector_type(2))) float v2f;
typedef __attribute__((ext_vector_type(8))) float v8f;

__device__ __forceinline__ float tanh_fast(float x) {
#if __has_builtin(__builtin_amdgcn_tanhf)
    return __builtin_amdgcn_tanhf(x);          // gfx1250 v_tanh_f32
#else
    float e = __expf(2.0f * x);                // v_exp_f32 fallback
    return (e - 1.0f) / (e + 1.0f);
#endif
}

// ---------------------------------------------------------------------------
// Kernel 1: column softmax over param + bottom-up subtree accumulation.
// One thread owns one leaf column; output written K-major: psubT[leaf*640+node].
// Pad nodes 625..639 are zeroed so downstream WMMA needs no masking.
// ---------------------------------------------------------------------------
__global__ __launch_bounds__(256) void softmax_psub_kernel(
    const float* __restrict__ param, float* __restrict__ psubT) {
    const int c = blockIdx.x * blockDim.x + threadIdx.x;   // leaf column
    if (c >= N_LEAF) return;

    float mx = -3.402823e38f;
    for (int r = 0; r < N_INNER; ++r)
        mx = fmaxf(mx, param[r * N_LEAF + c]);

    float s = 0.0f;
    for (int r = 0; r < N_INNER; ++r)
        s += __expf(param[r * N_LEAF + c] - mx);
    const float inv = 1.0f / s;

    float* row = psubT + (size_t)c * N_INNER_P;            // this thread's row
    for (int r = 0; r < N_INNER; ++r)
        row[r] = __expf(param[r * N_LEAF + c] - mx) * inv; // B[r][c]
    for (int r = N_INNER; r < N_INNER_P; ++r)
        row[r] = 0.0f;

    // (I - D1)^{-1} @ B == subtree sums.  Children always have larger index,
    // so a single descending sweep finalizes every node.
    // Node 124 has 4 children {621..624}; nodes r<124 have {5r+1..5r+5}.
    row[124] += row[621] + row[622] + row[623] + row[624];
    for (int r = 123; r >= 0; --r) {
        row[r] += row[5*r + 1] + row[5*r + 2] + row[5*r + 3]
                + row[5*r + 4] + row[5*r + 5];
    }
}

// ---------------------------------------------------------------------------
// Kernel 2: massT[k][d] = mass[d][k] so WMMA A-fragment loads are coalesced.
// ---------------------------------------------------------------------------
__global__ __launch_bounds__(256) void transpose_mass_kernel(
    const float* __restrict__ mass, float* __restrict__ massT) {
    const int idx = blockIdx.x * blockDim.x + threadIdx.x; // 0 .. 128*4096-1
    const int d = idx & (N_DOC - 1);
    const int k = idx >> 7;
    massT[k * N_DOC + d] = mass[d * N_LEAF + k];
}

// ---------------------------------------------------------------------------
// Kernel 3: z = mass @ psub.T via V_WMMA_F32_16X16X4_F32.
// One wave per 16x16 tile of z (8 M-tiles x 40 N-tiles).  K = 4096.
//
// A frag (16x4, MxK): lanes 0-15 hold row M=lane with {K=k0, k0+1} in v[0..1];
//                     lanes 16-31 hold {K=k0+2, k0+3}.
// B frag (4x16, KxN): lanes 0-15 hold col N=lane with rows {k0, k0+1};
//                     lanes 16-31 hold rows {k0+2, k0+3}.
// Both arrays are stored K-major so the 16-lane groups read contiguous floats.
// ---------------------------------------------------------------------------
__global__ __launch_bounds__(32) void gemm_z_kernel(
    const float* __restrict__ massT,   // (4096, 128)
    const float* __restrict__ psubT,   // (4096, 640)
    float* __restrict__ z) {           // (128, 640)
    const int nt   = blockIdx.x % (N_INNER_P / 16);
    const int mt   = blockIdx.x / (N_INNER_P / 16);
    const int lane = threadIdx.x;
    const int half = lane >> 4;        // 0: K pair {0,1}, 1: K pair {2,3}
    const int lp   = lane & 15;
    const int m0   = mt * 16;
    const int n0   = nt * 16;

    const float* Ab = massT + (m0 + lp);   // + k*128
    const float* Bb = psubT + (n0 + lp);   // + k*640

    v8f c = {};
#pragma unroll 8
    for (int k0 = 0; k0 < N_LEAF; k0 += 4) {
        const int ka = k0 + 2 * half;
        v2f a, b;
        a.x = Ab[(size_t)ka * N_DOC];
        a.y = Ab[(size_t)(ka + 1) * N_DOC];
        b.x = Bb[(size_t)ka * N_INNER_P];
        b.y = Bb[(size_t)(ka + 1) * N_INNER_P];
        // D = A x B + C  (f32, 16x16x4)
        c = __builtin_amdgcn_wmma_f32_16x16x4_f32(
            /*neg_a=*/false, a, /*neg_b=*/false, b,
            /*c_mod=*/(short)0, c, /*reuse_a=*/false, /*reuse_b=*/false);
    }

    // C/D layout: VGPR v holds M=v (lanes 0-15) and M=v+8 (lanes 16-31).
    float* zp = z + n0 + lp;
#pragma unroll
    for (int v = 0; v < 8; ++v)
        zp[(size_t)(m0 + v + 8 * half) * N_INNER_P] = c[v];
}

// ---------------------------------------------------------------------------
// Kernel 4: pairwise smoothabs distances.
// 8x8 pair tiles, 64 threads (2 waves) per workgroup, 256 workgroups.
// Features staged through LDS in 128-float chunks (stride-padded).
// Pad z-features (625..639) are zero for every doc -> contribute 0.
// ---------------------------------------------------------------------------
#define CHUNK 128
#define LDSTR (CHUNK + 4)

__global__ __launch_bounds__(64) void pairwise_kernel(
    const float* __restrict__ z,      // (128, 640)
    const float* __restrict__ mass,   // (128, 4096)
    float* __restrict__ out) {        // (128, 128)
    __shared__ float sa[8][LDSTR];
    __shared__ float sb[8][LDSTR];

    const int bt  = blockIdx.x & 15;
    const int at  = blockIdx.x >> 4;
    const int tid = threadIdx.x;      // 0..63
    const int al  = tid >> 3;         // 0..7
    const int bl  = tid & 7;          // 0..7

    float acc = 0.0f;

    // ---- inner-node distances over z (640 features incl. zero pad) ----
    for (int f0 = 0; f0 < N_INNER_P; f0 += CHUNK) {
        for (int i = tid; i < 8 * CHUNK; i += 64) {
            const int r = i >> 7, f = i & (CHUNK - 1);
            sa[r][f] = z[(size_t)(at * 8 + r) * N_INNER_P + f0 + f];
            sb[r][f] = z[(size_t)(bt * 8 + r) * N_INNER_P + f0 + f];
        }
        __syncthreads();
#pragma unroll 4
        for (int f = 0; f < CHUNK; ++f) {
            const float x = sa[al][f] - sb[bl][f];
            acc += x * tanh_fast(x);  // smoothabs(x, 2) == x*tanh(x)
        }
        __syncthreads();
    }

    // ---- leaf distances over mass (4096 features) ----
    for (int f0 = 0; f0 < N_LEAF; f0 += CHUNK) {
        for (int i = tid; i < 8 * CHUNK; i += 64) {
            const int r = i >> 7, f = i & (CHUNK - 1);
            sa[r][f] = mass[(size_t)(at * 8 + r) * N_LEAF + f0 + f];
            sb[r][f] = mass[(size_t)(bt * 8 + r) * N_LEAF + f0 + f];
        }
        __syncthreads();
#pragma unroll 4
        for (int f = 0; f < CHUNK; ++f) {
            const float x = sa[al][f] - sb[bl][f];
            acc += x * tanh_fast(x);
        }
        __syncthreads();
    }

    const int a = at * 8 + al;
    const int b = bt * 8 + bl;
    out[a * N_DOC + b] = acc;
}

// ---------------------------------------------------------------------------
extern "C" void kernel_launch(void* const* d_in, const int* in_sizes, int n_in,
                              void* d_out, int out_size, void* d_ws, size_t ws_size,
                              hipStream_t stream) {
    (void)in_sizes; (void)n_in; (void)out_size; (void)ws_size;

    const float* mass  = (const float*)d_in[0];   // (128, 4096)
    const float* param = (const float*)d_in[1];   // (625, 4096)
    float* out = (float*)d_out;                   // (128, 128)

    char* ws = (char*)d_ws;
    float* psubT = (float*)ws;                                     // 4096*640*4 = 10.49 MB
    float* massT = (float*)(ws + (size_t)N_LEAF * N_INNER_P * 4);  // 4096*128*4 =  2.10 MB
    float* zbuf  = (float*)(ws + (size_t)N_LEAF * N_INNER_P * 4
                               + (size_t)N_LEAF * N_DOC * 4);      //  128*640*4 =  0.33 MB

    softmax_psub_kernel<<<N_LEAF / 256, 256, 0, stream>>>(param, psubT);
    transpose_mass_kernel<<<(N_DOC * N_LEAF) / 256, 256, 0, stream>>>(mass, massT);
    gemm_z_kernel<<<(N_DOC / 16) * (N_INNER_P / 16), 32, 0, stream>>>(massT, psubT, zbuf);
    pairwise_kernel<<<(N_DOC / 8) * (N_DOC / 8), 64, 0, stream>>>(zbuf, mass, out);
}